// TGAE_Encoder_18923625906187
// MI455X (gfx1250) — compile-verified
//
#include <hip/hip_runtime.h>

typedef __attribute__((ext_vector_type(2))) float v2f;
typedef __attribute__((ext_vector_type(8))) float v8f;

#define DFEAT 128
#define KPAIRS 64                 // 128 K values stored as 64 float2 pair-rows
#define PITCH  144                // pairs per pair-row; 144*2 % 64 == 32 dwords
                                  // -> half-wave groups hit disjoint bank halves
#define TPB   256                 // 8 waves per block (wave32)
#define WPB   (TPB / 32)

// ---------------------------------------------------------------------------
// Generic 128-K GEMM: out[tile rows, 0..127] (+=) A @ W (+ bias), optional
// ReLU, optional duplicate store to out2 (used to seed agg = h).
// One wave computes a 16x128 tile via 32 k-steps x 8 col-tiles of
// v_wmma_f32_16x16x4_f32. W lives in LDS in k-pair-interleaved layout so each
// B fragment is a single aligned ds_load_b64. Flags are compile-time.
// ---------------------------------------------------------------------------
template <bool RELU, bool ACCUM, bool BIAS, bool OUT2>
__global__ __launch_bounds__(TPB) void k_gemm128(
    const float* __restrict__ A, const float* __restrict__ W,
    const float* __restrict__ bias, float* __restrict__ out,
    float* __restrict__ out2, int ntiles) {
  extern __shared__ float ldsf[];           // KPAIRS * PITCH float2 pairs
  v2f* lds = (v2f*)ldsf;
  // lds[pr * PITCH + n] = { W[2*pr][n], W[2*pr+1][n] }
  for (int i = threadIdx.x; i < KPAIRS * DFEAT; i += blockDim.x) {
    const int pr = i >> 7;
    const int n  = i & 127;
    v2f p;
    p.x = W[(size_t)(2 * pr) * DFEAT + n];
    p.y = W[(size_t)(2 * pr + 1) * DFEAT + n];
    lds[pr * PITCH + n] = p;
  }
  __syncthreads();

  const int lane = threadIdx.x & 31;
  const int wave = threadIdx.x >> 5;
  const int m    = lane & 15;      // row (A) / col (B,C) within tile
  const int half = lane >> 4;
  const int koff = half << 1;      // K sub-offset (even) for A/B fragments

  // Hoist per-lane bias values (column-indexed, tile-invariant).
  float bv[8];
#pragma unroll
  for (int ct = 0; ct < 8; ++ct) bv[ct] = BIAS ? bias[ct * 16 + m] : 0.0f;

  for (int tile = blockIdx.x * WPB + wave; tile < ntiles;
       tile += gridDim.x * WPB) {
    const int row0 = tile << 4;

    v8f acc[8];
    if (ACCUM) {
#pragma unroll
      for (int ct = 0; ct < 8; ++ct) {
        const int col = ct * 16 + m;
#pragma unroll
        for (int r = 0; r < 8; ++r)
          acc[ct][r] = out[(size_t)(row0 + r + 8 * half) * DFEAT + col];
      }
    } else {
#pragma unroll
      for (int ct = 0; ct < 8; ++ct)
#pragma unroll
        for (int r = 0; r < 8; ++r) acc[ct][r] = 0.0f;
    }

    const float* aRow = A + (size_t)(row0 + m) * DFEAT;
#pragma unroll 2
    for (int k0 = 0; k0 < DFEAT; k0 += 4) {
      const int kk = k0 + koff;                      // even
      // A fragment: lane l -> A[row0 + (l&15)][kk .. kk+1]
      v2f a = *(const v2f*)(aRow + kk);
      const v2f* bRow = lds + (kk >> 1) * PITCH + m;
#pragma unroll
      for (int ct = 0; ct < 8; ++ct) {
        // B fragment: lane l -> { W[kk][ct*16 + (l&15)], W[kk+1][...] }
        v2f b = bRow[ct * 16];
        acc[ct] = __builtin_amdgcn_wmma_f32_16x16x4_f32(
            false, a, false, b, (short)0, acc[ct], false, false);
      }
    }

#pragma unroll
    for (int ct = 0; ct < 8; ++ct) {
      const int col = ct * 16 + m;
#pragma unroll
      for (int r = 0; r < 8; ++r) {
        const int row = row0 + r + 8 * half;
        float v = acc[ct][r] + bv[ct];
        if (RELU) v = fmaxf(v, 0.0f);
        out[(size_t)row * DFEAT + col] = v;
        if (OUT2) out2[(size_t)row * DFEAT + col] = v;
      }
    }
  }
}

// ---------------------------------------------------------------------------
// Edge scatter: agg[dst[e]][:] += h[src[e]][:]  (agg pre-seeded with h -> GIN
// self term, eps = 0). One wave per edge; lane l handles floats 4l..4l+3.
// ---------------------------------------------------------------------------
__global__ __launch_bounds__(TPB) void k_scatter(
    const float* __restrict__ h, const int* __restrict__ src,
    const int* __restrict__ dst, float* __restrict__ agg, int E) {
  const int gw   = (int)((blockIdx.x * blockDim.x + threadIdx.x) >> 5);
  const int nw   = (int)((gridDim.x * blockDim.x) >> 5);
  const int lane = threadIdx.x & 31;
  for (int e = gw; e < E; e += nw) {
    const int s = src[e];
    const int d = dst[e];
    const float4 v = ((const float4*)(h + (size_t)s * DFEAT))[lane];
    float* ap = agg + (size_t)d * DFEAT + lane * 4;
    atomicAdd(ap + 0, v.x);
    atomicAdd(ap + 1, v.y);
    atomicAdd(ap + 2, v.z);
    atomicAdd(ap + 3, v.w);
  }
}

extern "C" void kernel_launch(void* const* d_in, const int* in_sizes, int n_in,
                              void* d_out, int out_size, void* d_ws,
                              size_t ws_size, hipStream_t stream) {
  const float* x     = (const float*)d_in[0];
  const int*   ei    = (const int*)d_in[1];   // int32 (JAX canonicalizes int64)
  const float* W_in  = (const float*)d_in[2];
  const float* b_in  = (const float*)d_in[3];
  const float* W1[3] = {(const float*)d_in[4], (const float*)d_in[8],
                        (const float*)d_in[12]};
  const float* b1[3] = {(const float*)d_in[5], (const float*)d_in[9],
                        (const float*)d_in[13]};
  const float* W2[3] = {(const float*)d_in[6], (const float*)d_in[10],
                        (const float*)d_in[14]};
  const float* b2[3] = {(const float*)d_in[7], (const float*)d_in[11],
                        (const float*)d_in[15]};
  const float* W_out = (const float*)d_in[16];
  const float* b_out = (const float*)d_in[17];
  float* out = (float*)d_out;

  const int N = in_sizes[0] / DFEAT;   // 100000
  const int E = in_sizes[1] / 2;       // 1600000
  const int* src = ei;
  const int* dst = ei + E;

  const size_t per = (size_t)N * DFEAT;
  float* base = (float*)d_ws;
  float* h[4];
  for (int j = 0; j < 4; ++j) h[j] = base + j * per;
  float* agg = base + 4 * per;

  const int ntiles = N >> 4;  // N is a multiple of 16
  const int gemmGrid = (ntiles + WPB - 1) / WPB;
  const int scatGrid = (E + WPB - 1) / WPB;
  const size_t ldsBytes = (size_t)KPAIRS * PITCH * sizeof(v2f);  // 72 KB

  // in_proj: h0 = x @ W_in + b_in, and seed agg = h0
  k_gemm128<false, false, true, true>
      <<<gemmGrid, TPB, ldsBytes, stream>>>(x, W_in, b_in, h[0], agg, ntiles);

  for (int i = 0; i < 3; ++i) {
    // agg (== h_i) += sum of neighbor features
    k_scatter<<<scatGrid, TPB, 0, stream>>>(h[i], src, dst, agg, E);
    // y = relu(agg @ W1 + b1)   (y stored into h[i+1]'s buffer)
    k_gemm128<true, false, true, false><<<gemmGrid, TPB, ldsBytes, stream>>>(
        agg, W1[i], b1[i], h[i + 1], nullptr, ntiles);
    // h_{i+1} = y @ W2 + b2 (in place, row-tile safe); seed agg for next layer
    if (i < 2) {
      k_gemm128<false, false, true, true><<<gemmGrid, TPB, ldsBytes, stream>>>(
          h[i + 1], W2[i], b2[i], h[i + 1], agg, ntiles);
    } else {
      k_gemm128<false, false, true, false>
          <<<gemmGrid, TPB, ldsBytes, stream>>>(h[i + 1], W2[i], b2[i],
                                                h[i + 1], nullptr, ntiles);
    }
  }

  // out = concat(h0..h3) @ W_out + b_out  ==  sum_j h_j @ W_out[128j:128j+128]
  k_gemm128<false, false, true, false><<<gemmGrid, TPB, ldsBytes, stream>>>(
      h[0], W_out, b_out, out, nullptr, ntiles);
  for (int j = 1; j < 4; ++j) {
    k_gemm128<false, true, false, false><<<gemmGrid, TPB, ldsBytes, stream>>>(
        h[j], W_out + (size_t)j * DFEAT * DFEAT, nullptr, out, nullptr,
        ntiles);
  }
}